// ComplexElementDependentResidualInteractionBlock_8675833938670
// MI455X (gfx1250) — compile-verified
//
#include <hip/hip_runtime.h>
#include <hip/hip_bf16.h>

typedef float v2f __attribute__((ext_vector_type(2)));
typedef float v8f __attribute__((ext_vector_type(8)));

#define NNODES 16000
#define NEDGES 256000
#define CH 64
#define NA 10
#define NR 8
#define NM 9

// ---------------------------------------------------------------------------
// Zero the scatter accumulators (needed every call: atomics accumulate).
// ---------------------------------------------------------------------------
__global__ void zero_kernel(float4* __restrict__ p, int n4) {
    int i = blockIdx.x * blockDim.x + threadIdx.x;
    if (i < n4) p[i] = make_float4(0.f, 0.f, 0.f, 0.f);
}

// ---------------------------------------------------------------------------
// sc[n,k] = na[n,a]*sum_u nf[n,u]*W_skip[u,a,k] / sqrt(C*A)   (one-hot attrs)
// Also extracts integer species per node for the edge kernel.
// ---------------------------------------------------------------------------
__global__ void sc_species_kernel(const float* __restrict__ na,
                                  const float* __restrict__ nf,
                                  const float* __restrict__ W_skip,
                                  float* __restrict__ sc_out,
                                  int* __restrict__ species) {
    int idx = blockIdx.x * blockDim.x + threadIdx.x;  // n*CH + k
    if (idx >= NNODES * CH) return;
    int n = idx >> 6;
    int k = idx & 63;
    // one-hot argmax (faithful for one-hot node_attrs)
    int a = 0;
    float best = na[n * NA];
#pragma unroll
    for (int j = 1; j < NA; ++j) {
        float v = na[n * NA + j];
        if (v > best) { best = v; a = j; }
    }
    if (k == 0) species[n] = a;
    float acc = 0.f;
#pragma unroll 8
    for (int u = 0; u < CH; ++u)
        acc += nf[n * CH + u] * W_skip[(u * NA + a) * CH + k];
    sc_out[idx] = acc * best * 0.03952847075210474f;  // 1/sqrt(640)
}

// ---------------------------------------------------------------------------
// h = nf @ W_up / sqrt(C) : N x 64 x 64 fp32 GEMM via V_WMMA_F32_16X16X4_F32.
// One wave per 16x16 output tile; 16 chained WMMAs over K=64.
// A layout: lane M = lane&15, K = vgpr + 2*(lane>>4)
// B layout: lane N = lane&15, K = vgpr + 2*(lane>>4)
// D layout: vgpr r -> M = r + 8*(lane>>4), N = lane&15
// ---------------------------------------------------------------------------
__global__ void linear_up_wmma(const float* __restrict__ nf,
                               const float* __restrict__ W_up,
                               float* __restrict__ h) {
    int wave = (blockIdx.x * blockDim.x + threadIdx.x) >> 5;
    int lane = threadIdx.x & 31;
    if (wave >= (NNODES / 16) * (CH / 16)) return;  // wave-uniform
    int vTile = wave & 3;
    int nTile = wave >> 2;
    int n0 = nTile * 16, v0 = vTile * 16;
    int row = lane & 15, hh = lane >> 4;
    v8f c = {};
    for (int k0 = 0; k0 < CH; k0 += 4) {
        int ka = k0 + 2 * hh;
        v2f a, b;
        a.x = nf[(n0 + row) * CH + ka];
        a.y = nf[(n0 + row) * CH + ka + 1];
        b.x = W_up[ka * CH + (v0 + row)];
        b.y = W_up[(ka + 1) * CH + (v0 + row)];
        c = __builtin_amdgcn_wmma_f32_16x16x4_f32(false, a, false, b,
                                                  (short)0, c, false, false);
    }
    const float s = 0.125f;  // 1/sqrt(64)
#pragma unroll
    for (int r = 0; r < 8; ++r)
        h[(n0 + r + 8 * hh) * CH + (v0 + row)] = c[r] * s;
}

// ---------------------------------------------------------------------------
// Per-edge: tp_w[l,u] = sum_r ef[e,r]*W_tpw[a,r,l,u]; then
// m[recv,u,m] += h[send,u]*tp_w[l(m),u]*Y[e,m] for real & imag.
// One wave per edge, 2 channels per lane. 74 MB accumulator lives in the
// 192 MB L2, so the 295M f32 atomics stay on-chip.
// ---------------------------------------------------------------------------
__global__ void edge_scatter_kernel(const float* __restrict__ ear,
                                    const float* __restrict__ eai,
                                    const float* __restrict__ ef,
                                    const int* __restrict__ senders,
                                    const int* __restrict__ receivers,
                                    const float* __restrict__ W_tpw,
                                    const float* __restrict__ h,
                                    const int* __restrict__ species,
                                    float* __restrict__ m_real,
                                    float* __restrict__ m_imag) {
    int e = (blockIdx.x * blockDim.x + threadIdx.x) >> 5;
    int lane = threadIdx.x & 31;
    if (e >= NEDGES) return;
    int s = senders[e];
    int r = receivers[e];
    int a = species[s];
    float rf[NR];
#pragma unroll
    for (int i = 0; i < NR; ++i) rf[i] = ef[e * NR + i];
    float yr[NM], yi[NM];
#pragma unroll
    for (int m = 0; m < NM; ++m) {
        yr[m] = ear[e * NM + m];
        yi[m] = eai[e * NM + m];
    }
#pragma unroll
    for (int half = 0; half < 2; ++half) {
        int u = lane + 32 * half;
        float x = h[s * CH + u];
        float tw0 = 0.f, tw1 = 0.f, tw2 = 0.f;
#pragma unroll
        for (int rr = 0; rr < NR; ++rr) {
            const float* wp = &W_tpw[((a * NR + rr) * 3) * CH + u];
            tw0 += rf[rr] * wp[0];
            tw1 += rf[rr] * wp[CH];
            tw2 += rf[rr] * wp[2 * CH];
        }
        float* pr = &m_real[(r * CH + u) * NM];
        float* pi = &m_imag[(r * CH + u) * NM];
#pragma unroll
        for (int m = 0; m < NM; ++m) {
            float tw = (m == 0) ? tw0 : ((m < 4) ? tw1 : tw2);
            float w = x * tw;
            atomicAdd(pr + m, w * yr[m]);
            atomicAdd(pi + m, w * yi[m]);
        }
    }
}

// ---------------------------------------------------------------------------
// out[n,v,m] = sum_u m[n,u,m]*W_lin[l(m)][u,v] / (sqrt(C)*AVG_NEIGH)
// 18 independent N x 64 x 64 fp32 GEMMs (9 m-components x real/imag),
// one wave per 16x16 tile, WMMA f32 16x16x4 chained over K=64.
// ---------------------------------------------------------------------------
__global__ void out_linear_wmma(const float* __restrict__ m_buf_real,
                                const float* __restrict__ m_buf_imag,
                                const float* __restrict__ W_lin,
                                float* __restrict__ out) {
    int wave = (blockIdx.x * blockDim.x + threadIdx.x) >> 5;
    int lane = threadIdx.x & 31;
    const int WAVES = 2 * (NNODES / 16) * (CH / 16) * NM;  // 72000
    if (wave >= WAVES) return;  // wave-uniform
    int m = wave % NM;
    int t = wave / NM;
    int vTile = t & 3;
    t >>= 2;
    int nTile = t % (NNODES / 16);
    int ri = t / (NNODES / 16);
    const float* mb = ri ? m_buf_imag : m_buf_real;
    float* ob = out + (size_t)ri * (size_t)(NNODES * CH * NM);
    int l = (m == 0) ? 0 : ((m < 4) ? 1 : 2);
    int n0 = nTile * 16, v0 = vTile * 16;
    int row = lane & 15, hh = lane >> 4;
    v8f c = {};
    for (int k0 = 0; k0 < CH; k0 += 4) {
        int ka = k0 + 2 * hh;
        v2f a, b;
        a.x = mb[((n0 + row) * CH + ka) * NM + m];
        a.y = mb[((n0 + row) * CH + ka + 1) * NM + m];
        b.x = W_lin[(l * CH + ka) * CH + (v0 + row)];
        b.y = W_lin[(l * CH + ka + 1) * CH + (v0 + row)];
        c = __builtin_amdgcn_wmma_f32_16x16x4_f32(false, a, false, b,
                                                  (short)0, c, false, false);
    }
    const float s = 0.0078125f;  // (1/sqrt(64)) * (1/16)
#pragma unroll
    for (int rr = 0; rr < 8; ++rr)
        ob[((n0 + rr + 8 * hh) * CH + (v0 + row)) * NM + m] = c[rr] * s;
}

// ---------------------------------------------------------------------------
extern "C" void kernel_launch(void* const* d_in, const int* in_sizes, int n_in,
                              void* d_out, int out_size, void* d_ws, size_t ws_size,
                              hipStream_t stream) {
    const float* node_attrs = (const float*)d_in[0];
    const float* node_feats = (const float*)d_in[1];
    const float* ear        = (const float*)d_in[2];
    const float* eai        = (const float*)d_in[3];
    const float* ef         = (const float*)d_in[4];
    const int*   senders    = (const int*)d_in[5];
    const int*   receivers  = (const int*)d_in[6];
    const float* W_up       = (const float*)d_in[7];
    const float* W_tpw      = (const float*)d_in[8];
    const float* W_lin      = (const float*)d_in[9];
    const float* W_skip     = (const float*)d_in[10];

    float* out = (float*)d_out;
    float* ws  = (float*)d_ws;

    float* h      = ws;                              // N*C floats
    float* m_real = ws + (size_t)NNODES * CH;        // N*C*9 floats
    float* m_imag = m_real + (size_t)NNODES * CH * NM;
    int*   species = (int*)(m_imag + (size_t)NNODES * CH * NM);

    // 1) zero scatter accumulators (every call — atomics accumulate)
    int n4 = (2 * NNODES * CH * NM) / 4;  // 4,608,000 float4s
    zero_kernel<<<(n4 + 255) / 256, 256, 0, stream>>>((float4*)m_real, n4);

    // 2) skip connection + species extraction
    sc_species_kernel<<<(NNODES * CH + 255) / 256, 256, 0, stream>>>(
        node_attrs, node_feats, W_skip,
        out + (size_t)2 * NNODES * CH * NM, species);

    // 3) h = nf @ W_up / sqrt(C)  (WMMA)  : 4000 waves -> 500 blocks
    linear_up_wmma<<<500, 256, 0, stream>>>(node_feats, W_up, h);

    // 4) fused tp_w + outer product + atomic scatter-sum (one wave per edge)
    edge_scatter_kernel<<<NEDGES / 8, 256, 0, stream>>>(
        ear, eai, ef, senders, receivers, W_tpw, h, species, m_real, m_imag);

    // 5) 18 output GEMMs (WMMA) : 72000 waves -> 9000 blocks
    out_linear_wmma<<<9000, 256, 0, stream>>>(m_real, m_imag, W_lin, out);
}